// ConfusionAttentionModule_73632919323399
// MI455X (gfx1250) — compile-verified
//
#include <hip/hip_runtime.h>

// ---------------------------------------------------------------------------
// score[b] = (sum of src rows in seg b) . (sum of tar rows in seg b)
//            / (cnt_src[b] * cnt_tar[b])
//
// MI455X roofline: mandatory traffic = 2 * 200000 * 256 * 4B = 410 MB read
// once => ~17.6us at 23.3 TB/s.  FLOPs (~0.1 GFLOP) are irrelevant, so the
// kernel is designed purely around streaming bandwidth:
//   - phase 1: contiguous 128-bit non-temporal loads, register accumulation
//     using the fact that batch ids are sorted (flush to global only at
//     segment boundaries -> a few global_atomic_add_f32 per block).
//   - phase 2: counts via binary search (no atomics at all).
//   - phase 3: per-segment dot product on V_WMMA_F32_16X16X4_F32 (the only
//     matmul-shaped piece), diagonal extracted per the CDNA5 C-layout.
// ---------------------------------------------------------------------------

typedef __attribute__((ext_vector_type(2))) float v2f;
typedef __attribute__((ext_vector_type(4))) float v4f;
typedef __attribute__((ext_vector_type(8))) float v8f;

#define RPB 256  // rows per block in phase 1

__global__ void zero_ws_kernel(float* __restrict__ p, int n) {
  int i = blockIdx.x * blockDim.x + threadIdx.x;
  if (i < n) p[i] = 0.0f;
}

// Phase 1: segmented row-sum over sorted ids.
// blockDim.x = D/4 (=64): thread t owns columns [4t, 4t+4) as one float4.
// A wave of 32 lanes issues one contiguous 512B global_load_b128 per row.
__global__ void seg_rowsum_kernel(const float* __restrict__ x,
                                  const int* __restrict__ batch,
                                  float* __restrict__ sums,
                                  int n_rows, int D) {
  __shared__ int sb[RPB];
  const int r0 = blockIdx.x * RPB;
  int nr = n_rows - r0;
  if (nr > RPB) nr = RPB;
  for (int i = threadIdx.x; i < nr; i += blockDim.x) sb[i] = batch[r0 + i];
  __syncthreads();

  const int c4 = threadIdx.x;                 // which float4 of the row
  const v4f* xp = (const v4f*)x;
  const size_t d4 = (size_t)(D >> 2);

  v4f acc = {0.0f, 0.0f, 0.0f, 0.0f};
  int cur = sb[0];
  for (int i = 0; i < nr; ++i) {
    // streaming read: never reused -> non-temporal hint
    v4f v = __builtin_nontemporal_load(&xp[(size_t)(r0 + i) * d4 + c4]);
    int seg = sb[i];                          // uniform across the block
    if (seg != cur) {                         // segment boundary: flush
      float* dst = sums + (size_t)cur * D + (c4 << 2);
      atomicAdd(dst + 0, acc.x);
      atomicAdd(dst + 1, acc.y);
      atomicAdd(dst + 2, acc.z);
      atomicAdd(dst + 3, acc.w);
      cur = seg;
      acc = v;
    } else {
      acc += v;
    }
  }
  float* dst = sums + (size_t)cur * D + (c4 << 2);
  atomicAdd(dst + 0, acc.x);
  atomicAdd(dst + 1, acc.y);
  atomicAdd(dst + 2, acc.z);
  atomicAdd(dst + 3, acc.w);
}

// Phase 2: per-segment counts via binary search on the sorted id arrays.
__device__ static inline int lower_bound_i32(const int* __restrict__ a, int n, int key) {
  int lo = 0, hi = n;
  while (lo < hi) {
    int mid = (lo + hi) >> 1;
    if (a[mid] < key) lo = mid + 1; else hi = mid;
  }
  return lo;
}

__global__ void seg_count_kernel(const int* __restrict__ bs, int ns,
                                 const int* __restrict__ bt, int nt,
                                 float* __restrict__ cs, float* __restrict__ ct, int B) {
  int b = blockIdx.x * blockDim.x + threadIdx.x;
  if (b >= B) return;
  cs[b] = (float)(lower_bound_i32(bs, ns, b + 1) - lower_bound_i32(bs, ns, b));
  ct[b] = (float)(lower_bound_i32(bt, nt, b + 1) - lower_bound_i32(bt, nt, b));
}

// Phase 3: score[b] = dot(S[b,:], T[b,:]) / (cs[b]*ct[b]) for 16 segments per
// wave via V_WMMA_F32_16X16X4_F32:  C(16x16) += A(16x4) * B(4x16) with
//   A[m,k] = S[sg+m, k0+k],  B[k,n] = T[sg+n, k0+k]   => C[m,n] = dot over K.
// We only keep the diagonal C[m,m].
// A layout (32-bit 16x4): lanes 0-15 hold M=lane (K=0,1 in v0,v1),
//                         lanes 16-31 hold M=lane-16 (K=2,3).
// B layout (32-bit 4x16): lanes 0-15 hold N=lane (K=0 in v0, K=1 in v1),
//                         lanes 16-31 hold N=lane-16 (K=2,3).
// -> per lane: same k-pair offset (l>=16 ? +2 : +0) for both operands.
// C layout (32-bit 16x16): VGPR v, lanes 0-15: M=v, N=lane;
//                          lanes 16-31: M=v+8, N=lane-16.
// Diagonal: M<8 -> lane=M, elem M;  M>=8 -> lane=M+16, elem M-8.
__global__ void wmma_seg_dot_kernel(const float* __restrict__ S,
                                    const float* __restrict__ T,
                                    const float* __restrict__ cs,
                                    const float* __restrict__ ct,
                                    float* __restrict__ out, int D) {
  const int l  = threadIdx.x;        // 0..31, one full wave, EXEC all ones
  const int sg = blockIdx.x * 16;    // 16 segments per block
  const int m16  = l & 15;
  const int koff = (l >> 4) << 1;    // 0 for lanes 0-15, 2 for lanes 16-31

  const float* arow = S + (size_t)(sg + m16) * D;
  const float* brow = T + (size_t)(sg + m16) * D;

  v8f c = {0.f, 0.f, 0.f, 0.f, 0.f, 0.f, 0.f, 0.f};
  for (int k0 = 0; k0 < D; k0 += 4) {
    const int kb = k0 + koff;
    v2f a; a.x = arow[kb]; a.y = arow[kb + 1];
    v2f b; b.x = brow[kb]; b.y = brow[kb + 1];
    // (neg_a, A, neg_b, B, c_mod, C, reuse_a, reuse_b)
    c = __builtin_amdgcn_wmma_f32_16x16x4_f32(false, a, false, b,
                                              (short)0, c, false, false);
  }

  union { v8f v; float f[8]; } u;
  u.v = c;
  int m = -1;
  float val = 0.0f;
  if (l < 8)        { m = l;      val = u.f[l]; }
  else if (l >= 24) { m = l - 16; val = u.f[l - 24]; }
  if (m >= 0) {
    const int b = sg + m;
    out[b] = val / (cs[b] * ct[b]);
  }
}

extern "C" void kernel_launch(void* const* d_in, const int* in_sizes, int n_in,
                              void* d_out, int out_size, void* d_ws, size_t ws_size,
                              hipStream_t stream) {
  const float* x_src     = (const float*)d_in[0];
  const int*   batch_src = (const int*)d_in[1];
  const float* x_tar     = (const float*)d_in[2];
  const int*   batch_tar = (const int*)d_in[3];

  const int n_src = in_sizes[1];
  const int n_tar = in_sizes[3];
  const int D     = in_sizes[0] / n_src;   // 256
  const int B     = out_size;              // 512

  // workspace: sum_src[B*D] | sum_tar[B*D] | cnt_src[B] | cnt_tar[B]
  float* sumS = (float*)d_ws;
  float* sumT = sumS + (size_t)B * D;
  float* cs   = sumT + (size_t)B * D;
  float* ct   = cs + B;

  const int zn = 2 * B * D + 2 * B;
  zero_ws_kernel<<<(zn + 255) / 256, 256, 0, stream>>>(sumS, zn);

  const int tpb = D / 4;  // 64 threads: one float4 column strip each
  seg_rowsum_kernel<<<(n_src + RPB - 1) / RPB, tpb, 0, stream>>>(
      x_src, batch_src, sumS, n_src, D);
  seg_rowsum_kernel<<<(n_tar + RPB - 1) / RPB, tpb, 0, stream>>>(
      x_tar, batch_tar, sumT, n_tar, D);

  seg_count_kernel<<<(B + 255) / 256, 256, 0, stream>>>(
      batch_src, n_src, batch_tar, n_tar, cs, ct, B);

  wmma_seg_dot_kernel<<<B / 16, 32, 0, stream>>>(sumS, sumT, cs, ct,
                                                 (float*)d_out, D);
}